// Magnus0_50422916055248
// MI455X (gfx1250) — compile-verified
//
#include <hip/hip_runtime.h>
#include <math.h>

typedef float v2f __attribute__((ext_vector_type(2)));
typedef float v8f __attribute__((ext_vector_type(8)));

// 2x2 row-major matrix in float4: x=m00 y=m01 z=m10 w=m11
__device__ __forceinline__ float4 mul2(const float4 A, const float4 B) {
    float4 r;
    r.x = fmaf(A.x, B.x, A.y * B.z);
    r.y = fmaf(A.x, B.y, A.y * B.w);
    r.z = fmaf(A.z, B.x, A.w * B.z);
    r.w = fmaf(A.z, B.y, A.w * B.w);
    return r;
}

// ---------------- Stage A: per-step closed-form 2x2 matrix exponential ----------------
__global__ void expm_kernel(const float* __restrict__ t,
                            const float* __restrict__ oms_p,
                            const float* __restrict__ g0_p,
                            const float* __restrict__ gamp_p,
                            const float* __restrict__ omd_p,
                            float4* __restrict__ E, int S) {
    int i = blockIdx.x * blockDim.x + threadIdx.x;
    if (i >= S) return;
    float t0 = t[i], t1 = t[i + 1];
    float dt = t1 - t0;
    float tm = t0 + 0.5f * dt;
    float gamma = g0_p[0] * (1.0f + gamp_p[0] * sinf(omd_p[0] * tm));
    // M = dt*A(tm) = [[0, dt], [-omega_sq*dt, -gamma*dt]]
    float a = 0.0f, b = dt, c = -oms_p[0] * dt, d = -gamma * dt;
    float m = 0.5f * (a + d);
    float delta = m * m - (a * d - b * c);
    float s = sqrtf(fabsf(delta));
    float ss = fmaxf(s, 1e-12f);
    bool pos = delta >= 0.0f;
    float f = pos ? coshf(s) : cosf(s);
    float g = (pos ? sinhf(ss) : sinf(ss)) / ss;
    float em = expf(m);
    float4 Ei;
    Ei.x = em * (f + g * (a - m));
    Ei.y = em * (g * b);
    Ei.z = em * (g * c);
    Ei.w = em * (f + g * (d - m));
    E[i] = Ei;
}

// ---------------- Stage B1: chunk-local left-prefix products (in place) ----------------
__global__ void chunk_scan_kernel(float4* __restrict__ P, int S, int CH) {
    int j = blockIdx.x * blockDim.x + threadIdx.x;
    int beg = j * CH;
    if (beg >= S) return;
    int end = min(beg + CH, S);
    float4 run = make_float4(1.f, 0.f, 0.f, 1.f);
    for (int i = beg; i < end; ++i) {
        run = mul2(P[i], run);   // P_i = E_i * P_{i-1} (left product)
        P[i] = run;
    }
}

// ---------------- Stage B2: exclusive scan of chunk totals (tiny, 1 thread) ----------------
__global__ void chunk_prefix_kernel(const float4* __restrict__ P, float4* __restrict__ C,
                                    int S, int CH, int NC) {
    if (threadIdx.x != 0 || blockIdx.x != 0) return;
    float4 acc = make_float4(1.f, 0.f, 0.f, 1.f);
    for (int j = 0; j < NC; ++j) {
        C[j] = acc;
        int last = min((j + 1) * CH, S) - 1;
        acc = mul2(P[last], acc);
    }
}

// ---------------- Stage B3: combine -> row matrix R[2T, 2] (rows 0,1 = identity) ----------
__global__ void build_rows_kernel(const float4* __restrict__ P, const float4* __restrict__ C,
                                  float* __restrict__ R, int S, int CH) {
    int i = blockIdx.x * blockDim.x + threadIdx.x;
    if (i >= S) return;
    float4 p = mul2(P[i], C[i / CH]);        // global prefix P_i
    int g = 2 * (i + 1);                     // output row for state i+1, component 0
    R[(size_t)g * 2 + 0] = p.x;  R[(size_t)g * 2 + 1] = p.y;
    R[(size_t)(g + 1) * 2 + 0] = p.z;  R[(size_t)(g + 1) * 2 + 1] = p.w;
    if (i == 0) { R[0] = 1.f; R[1] = 0.f; R[2] = 0.f; R[3] = 1.f; }
}

// ---------------- Stage C: out[16K, B] = R[16K, 2] @ x0[2, B] via V_WMMA_F32_16X16X4_F32 ----
// One wave per 16x16 output tile. K padded 2 -> 4 with zeros (lanes 16-31 hold K=2,3).
// Output is a 268 MB write-once stream (> 192 MB L2): store with non-temporal hint so the
// L2 stays hot for R (128 KB, re-read per column tile) and x0 (32 KB, re-read per row tile).
__global__ __launch_bounds__(256) void wmma_apply_kernel(const float* __restrict__ R,
                                                         const float* __restrict__ x0,
                                                         float* __restrict__ out,
                                                         int B, int numColTiles, int totalTiles) {
    int wave = blockIdx.x * (blockDim.x >> 5) + (threadIdx.x >> 5);
    if (wave >= totalTiles) return;          // wave-uniform exit, EXEC all-ones inside
    int lane = threadIdx.x & 31;
    int rowTile = wave / numColTiles;
    int colTile = wave % numColTiles;
    int l = lane & 15;
    bool lo = lane < 16;

    // A: 16x4 f32 layout -> lanes 0-15 carry K=0,1 for row M=lane; lanes 16-31 = K=2,3 pad = 0
    int rrow = rowTile * 16 + l;
    float a0 = R[(size_t)rrow * 2 + 0];
    float a1 = R[(size_t)rrow * 2 + 1];
    v2f a; a.x = lo ? a0 : 0.0f; a.y = lo ? a1 : 0.0f;

    // B: 4x16 f32 layout -> lanes 0-15 carry K=0,1 for col N=lane; lanes 16-31 = K=2,3 pad = 0
    int col = colTile * 16 + l;
    float b0 = x0[col];
    float b1 = x0[(size_t)B + col];
    v2f bb; bb.x = lo ? b0 : 0.0f; bb.y = lo ? b1 : 0.0f;

    v8f acc = {};
    acc = __builtin_amdgcn_wmma_f32_16x16x4_f32(false, a, false, bb, (short)0, acc,
                                                false, false);

    // D layout: VGPR r -> row (lanes 0-15: M=r, lanes 16-31: M=8+r), col = lane&15
    int mBase = rowTile * 16 + (lo ? 0 : 8);
    float* op = out + (size_t)mBase * B + (size_t)colTile * 16 + l;
#pragma unroll
    for (int r = 0; r < 8; ++r)
        __builtin_nontemporal_store(acc[r], op + (size_t)r * B);
}

extern "C" void kernel_launch(void* const* d_in, const int* in_sizes, int n_in,
                              void* d_out, int out_size, void* d_ws, size_t ws_size,
                              hipStream_t stream) {
    const float* t    = (const float*)d_in[0];
    const float* x0   = (const float*)d_in[1];
    const float* oms  = (const float*)d_in[2];
    const float* g0   = (const float*)d_in[3];
    const float* gamp = (const float*)d_in[4];
    const float* omd  = (const float*)d_in[5];
    float* out = (float*)d_out;

    const int T = in_sizes[0];          // 8192
    const int B = in_sizes[1] / 2;      // 4096
    const int S = T - 1;                // 8191 steps
    const int CH = 64;
    const int NC = (S + CH - 1) / CH;   // 128 chunks

    // Workspace layout (floats): E/Ploc [S*4] | Cpre [NC*4] | R [2T*2]
    float* ws = (float*)d_ws;
    float4* E    = (float4*)ws;                       // S float4 (reused in place)
    float4* Cpre = (float4*)(ws + (size_t)S * 4);     // NC float4
    float*  R    = ws + (size_t)S * 4 + (size_t)NC * 4;

    expm_kernel<<<(S + 255) / 256, 256, 0, stream>>>(t, oms, g0, gamp, omd, E, S);
    chunk_scan_kernel<<<(NC + 127) / 128, 128, 0, stream>>>(E, S, CH);
    chunk_prefix_kernel<<<1, 1, 0, stream>>>(E, Cpre, S, CH, NC);
    build_rows_kernel<<<(S + 255) / 256, 256, 0, stream>>>(E, Cpre, R, S, CH);

    const int rows = 2 * T;                    // 16384
    const int rowTiles = (rows + 15) / 16;     // 1024
    const int colTiles = B / 16;               // 256
    const int totalTiles = rowTiles * colTiles;
    const int wavesPerBlock = 256 / 32;        // 8
    const int blocks = (totalTiles + wavesPerBlock - 1) / wavesPerBlock;
    wmma_apply_kernel<<<blocks, 256, 0, stream>>>(R, x0, out, B, colTiles, totalTiles);
}